// VAR1_15985868276410
// MI455X (gfx1250) — compile-verified
//
#include <hip/hip_runtime.h>

// CDNA5 (gfx1250) chunked parallel-scan for z_t = A z_{t-1} + n_t, A = L^T L.
// fp32 WMMA (V_WMMA_F32_16X16X4_F32) keeps full precision over 4096 chained steps.
// Inner loop: dual-accumulator K-split (8-deep dependent WMMA chains) + next-step
// noise prefetch (global_prefetch_b8) to keep HBM latency off the serial path.

typedef __attribute__((ext_vector_type(2))) float v2f;
typedef __attribute__((ext_vector_type(8))) float v8f;

constexpr int kDim    = 64;
constexpr int kBatch  = 64;
constexpr int kSeq    = 4096;
constexpr int kChunk  = 64;              // timesteps per chunk
constexpr int kNChunk = kSeq / kChunk;   // 64
constexpr int kPitch  = kDim + 1;        // LDS pad: bank-conflict-free transposed reads

#define WMMA_F32(a, b, c) \
  __builtin_amdgcn_wmma_f32_16x16x4_f32(false, (a), false, (b), (short)0, (c), false, false)

// ---------------- Kernel 0: A = L^T L and A^64 by repeated squaring ----------
__global__ __launch_bounds__(256) void prep_kernel(const float* __restrict__ L,
                                                   float* __restrict__ Aout,
                                                   float* __restrict__ A64out) {
  __shared__ float sL[kDim * kDim];
  __shared__ float sA[kDim * kDim];
  __shared__ float sB[kDim * kDim];
  const int tid = threadIdx.x;
  for (int i = tid; i < kDim * kDim; i += 256) sL[i] = L[i];
  __syncthreads();
  for (int e = tid; e < kDim * kDim; e += 256) {
    const int i = e >> 6, j = e & 63;
    float acc = 0.f;
    for (int k = 0; k < kDim; ++k) acc = fmaf(sL[k * kDim + i], sL[k * kDim + j], acc);
    sA[e] = acc;
    Aout[e] = acc;
  }
  __syncthreads();
  float* src = sA;
  float* dst = sB;
  for (int s = 0; s < 6; ++s) {  // A^(2^6) = A^64
    for (int e = tid; e < kDim * kDim; e += 256) {
      const int i = e >> 6, j = e & 63;
      float acc = 0.f;
      for (int k = 0; k < kDim; ++k) acc = fmaf(src[i * kDim + k], src[k * kDim + j], acc);
      dst[e] = acc;
    }
    __syncthreads();
    float* t = src; src = dst; dst = t;
  }
  for (int e = tid; e < kDim * kDim; e += 256) A64out[e] = src[e];
}

// ------------- Phase 1 / Phase 3: run one chunk for one 16-row batch tile ----
// PHASE 0: zero initial state, emit only final state S_c.
// PHASE 1: initial state = Carry_c, emit every z_t to Zout.
template <int PHASE>
__global__ __launch_bounds__(32) void chunk_kernel(const float* __restrict__ noise,
                                                   const float* __restrict__ Amat,
                                                   const float* __restrict__ CarryIn,
                                                   float* __restrict__ Sout,
                                                   float* __restrict__ Zout) {
  __shared__ float Zs[16][kPitch];
  const int lane = threadIdx.x;
  const int lx   = lane & 15;
  const int half = lane >> 4;
  const int c    = blockIdx.x;        // chunk
  const int b0   = blockIdx.y * 16;   // batch tile
  const int t0   = c * kChunk;

  // Resident B fragments of A^T: B[k][n] = A[n][k]; (k = kk*4 + 2*half + v, n = 16*nn + lx)
  v2f bfrag[4][16];
#pragma unroll
  for (int nn = 0; nn < 4; ++nn)
#pragma unroll
    for (int kk = 0; kk < 16; ++kk) {
      const float* p = Amat + (nn * 16 + lx) * kDim + kk * 4 + 2 * half;
      v2f b; b.x = p[0]; b.y = p[1];
      bfrag[nn][kk] = b;
    }

  if (PHASE == 0) {
    for (int idx = lane; idx < 16 * kDim; idx += 32) Zs[idx >> 6][idx & 63] = 0.0f;
  } else {
    const float* cp = CarryIn + (size_t)c * (kBatch * kDim) + (size_t)b0 * kDim;
    for (int idx = lane; idx < 16 * kDim; idx += 32) Zs[idx >> 6][idx & 63] = cp[idx];
  }
  __syncthreads();

  const size_t rowStride = (size_t)kSeq * kDim;  // noise/out: [batch][seq][dim]

  for (int t = 0; t < kChunk; ++t) {
    const int tt = t0 + t;

    // Prefetch next step's 16x64 f32 noise tile: 64 cachelines -> 2 prefetches/wave.
    if (t + 1 < kChunk) {
#pragma unroll
      for (int i = 0; i < 2; ++i) {
        const int idx = lane + 32 * i;  // cacheline id: row = idx/4, 16-float column group = idx%4
        __builtin_prefetch(noise + (size_t)(b0 + (idx >> 2)) * rowStride +
                               (size_t)(tt + 1) * kDim + (idx & 3) * 16,
                           0, 1);
      }
    }

    // acc0 = N_t loaded directly in C/D fragment layout: (m = r + 8*half, n = 16*nn + lx)
    v8f acc0[4];
    v8f acc1[4] = {};
    const float* np = noise + (size_t)(b0 + 8 * half) * rowStride + (size_t)tt * kDim + lx;
#pragma unroll
    for (int nn = 0; nn < 4; ++nn)
#pragma unroll
      for (int r = 0; r < 8; ++r) acc0[nn][r] = np[(size_t)r * rowStride + nn * 16];

    // acc += Z @ A^T, K split across two accumulators to halve dependent-WMMA depth.
    // (A-frag: m = lx, k = kk*4 + 2*half + v ; LDS pad avoids bank conflicts)
#pragma unroll
    for (int kk = 0; kk < 8; ++kk) {
      v2f a0, a1;
      a0.x = Zs[lx][kk * 4 + 2 * half];
      a0.y = Zs[lx][kk * 4 + 2 * half + 1];
      a1.x = Zs[lx][(kk + 8) * 4 + 2 * half];
      a1.y = Zs[lx][(kk + 8) * 4 + 2 * half + 1];
#pragma unroll
      for (int nn = 0; nn < 4; ++nn) {
        acc0[nn] = WMMA_F32(a0, bfrag[nn][kk], acc0[nn]);
        acc1[nn] = WMMA_F32(a1, bfrag[nn][kk + 8], acc1[nn]);
      }
    }
#pragma unroll
    for (int nn = 0; nn < 4; ++nn) acc0[nn] = acc0[nn] + acc1[nn];

    __syncthreads();  // all lanes done reading old Z (1-wave WG: compiles to waits)
#pragma unroll
    for (int nn = 0; nn < 4; ++nn)
#pragma unroll
      for (int r = 0; r < 8; ++r) Zs[r + 8 * half][nn * 16 + lx] = acc0[nn][r];
    __syncthreads();

    if (PHASE == 1) {
      float* op = Zout + (size_t)(b0 + 8 * half) * rowStride + (size_t)tt * kDim + lx;
#pragma unroll
      for (int nn = 0; nn < 4; ++nn)
#pragma unroll
        for (int r = 0; r < 8; ++r) op[(size_t)r * rowStride + nn * 16] = acc0[nn][r];
    }
  }

  if (PHASE == 0) {
    float* sp = Sout + (size_t)c * (kBatch * kDim) + (size_t)b0 * kDim;
    for (int idx = lane; idx < 16 * kDim; idx += 32) sp[idx] = Zs[idx >> 6][idx & 63];
  }
}

// ------------- Phase 2: sequential carry combine: Carry' = Carry @ (A^64)^T + S_c ----
__global__ __launch_bounds__(32) void combine_kernel(const float* __restrict__ S,
                                                     const float* __restrict__ A64,
                                                     float* __restrict__ Carry) {
  __shared__ float Zs[16][kPitch];
  const int lane = threadIdx.x;
  const int lx   = lane & 15;
  const int half = lane >> 4;
  const int b0   = blockIdx.x * 16;

  v2f bfrag[4][16];
#pragma unroll
  for (int nn = 0; nn < 4; ++nn)
#pragma unroll
    for (int kk = 0; kk < 16; ++kk) {
      const float* p = A64 + (nn * 16 + lx) * kDim + kk * 4 + 2 * half;
      v2f b; b.x = p[0]; b.y = p[1];
      bfrag[nn][kk] = b;
    }

  for (int idx = lane; idx < 16 * kDim; idx += 32) Zs[idx >> 6][idx & 63] = 0.0f;  // Carry_0 = 0
  __syncthreads();

  for (int cidx = 0; cidx < kNChunk; ++cidx) {
    // Emit Carry_c = true state just before chunk c
    float* cp = Carry + (size_t)cidx * (kBatch * kDim) + (size_t)b0 * kDim;
    for (int idx = lane; idx < 16 * kDim; idx += 32) cp[idx] = Zs[idx >> 6][idx & 63];

    v8f acc0[4];
    v8f acc1[4] = {};
    const float* sp = S + (size_t)cidx * (kBatch * kDim);
#pragma unroll
    for (int nn = 0; nn < 4; ++nn)
#pragma unroll
      for (int r = 0; r < 8; ++r)
        acc0[nn][r] = sp[(size_t)(b0 + r + 8 * half) * kDim + nn * 16 + lx];

#pragma unroll
    for (int kk = 0; kk < 8; ++kk) {
      v2f a0, a1;
      a0.x = Zs[lx][kk * 4 + 2 * half];
      a0.y = Zs[lx][kk * 4 + 2 * half + 1];
      a1.x = Zs[lx][(kk + 8) * 4 + 2 * half];
      a1.y = Zs[lx][(kk + 8) * 4 + 2 * half + 1];
#pragma unroll
      for (int nn = 0; nn < 4; ++nn) {
        acc0[nn] = WMMA_F32(a0, bfrag[nn][kk], acc0[nn]);
        acc1[nn] = WMMA_F32(a1, bfrag[nn][kk + 8], acc1[nn]);
      }
    }
#pragma unroll
    for (int nn = 0; nn < 4; ++nn) acc0[nn] = acc0[nn] + acc1[nn];

    __syncthreads();
#pragma unroll
    for (int nn = 0; nn < 4; ++nn)
#pragma unroll
      for (int r = 0; r < 8; ++r) Zs[r + 8 * half][nn * 16 + lx] = acc0[nn][r];
    __syncthreads();
  }
}

extern "C" void kernel_launch(void* const* d_in, const int* in_sizes, int n_in,
                              void* d_out, int out_size, void* d_ws, size_t ws_size,
                              hipStream_t stream) {
  const float* noise = (const float*)d_in[0];  // [64][4096][64]
  const float* L     = (const float*)d_in[1];  // [64][64]
  float* out = (float*)d_out;

  // Workspace layout (~2.03 MB): A | A^64 | S (chunk end-states) | Carry (pre-chunk states)
  float* A     = (float*)d_ws;
  float* A64   = A + kDim * kDim;
  float* S     = A64 + kDim * kDim;
  float* Carry = S + (size_t)kNChunk * kBatch * kDim;

  prep_kernel<<<1, 256, 0, stream>>>(L, A, A64);
  chunk_kernel<0><<<dim3(kNChunk, kBatch / 16), 32, 0, stream>>>(noise, A, nullptr, S, nullptr);
  combine_kernel<<<kBatch / 16, 32, 0, stream>>>(S, A64, Carry);
  chunk_kernel<1><<<dim3(kNChunk, kBatch / 16), 32, 0, stream>>>(noise, A, Carry, nullptr, out);
}